// HybridQLSTM_65481071400200
// MI455X (gfx1250) — compile-verified
//
#include <hip/hip_runtime.h>
#include <math.h>

#define T_SEQ 4096
#define N_TAGS 32000
#define HID 4
#define NQ 4

typedef __attribute__((ext_vector_type(2))) float v2f;
typedef __attribute__((ext_vector_type(8))) float v8f;

// ---------------------------------------------------------------------------
// helpers
// ---------------------------------------------------------------------------
__device__ __forceinline__ float fast_sigmoid(float x) {
    return 1.0f / (1.0f + __expf(-x));
}
__device__ __forceinline__ float fast_tanh(float x) {
    // inputs here are bounded (|x| <~ 4), no overflow concern
    float e = __expf(2.0f * x);
    return (e - 1.0f) / (e + 1.0f);
}

// RX(theta) on one wire: state pair exchange across lanes. RX is symmetric,
// so both halves of the pair use the same update:
//   r' = c*r + s*partner_i ; i' = c*i - s*partner_r
__device__ __forceinline__ void rx_apply(float& r, float& i, float c, float s, int lmask) {
    float pr = __shfl_xor(r, lmask, 32);
    float pi = __shfl_xor(i, lmask, 32);
    float nr = __fmaf_rn(c, r,  s * pi);
    float ni = __fmaf_rn(c, i, -s * pr);
    r = nr; i = ni;
}

// CNOT(ctrl,tgt): amplitudes with ctrl bit set swap with target-bit partner
__device__ __forceinline__ void cnot_apply(float& r, float& i, int amp, int ctrl, int tgt) {
    float pr = __shfl_xor(r, 1 << tgt, 32);
    float pi = __shfl_xor(i, 1 << tgt, 32);
    if ((amp >> ctrl) & 1) { r = pr; i = pi; }
}

// sum across the 16-lane amplitude group (masks 1,2,4,8 stay inside half-wave)
__device__ __forceinline__ float half_reduce(float v) {
    v += __shfl_xor(v, 1, 32);
    v += __shfl_xor(v, 2, 32);
    v += __shfl_xor(v, 4, 32);
    v += __shfl_xor(v, 8, 32);
    return v;
}

// ---------------------------------------------------------------------------
// Kernel 1: 4096-step serial quantum-LSTM recurrence, single wave32.
// Lanes 0-15 hold amplitudes 0-15 of gates {0,2}; lanes 16-31 of gates {1,3}.
// Writes hx[t][0..3] (float4) to workspace.
// ---------------------------------------------------------------------------
__global__ __launch_bounds__(32) void qlstm_recurrence_kernel(
    const float* __restrict__ qcnn_w, const float* __restrict__ qcnn_b,
    const float* __restrict__ lin_w,  const float* __restrict__ lin_b,
    const float* __restrict__ rx_params, float* __restrict__ hx_out)
{
    const int l   = threadIdx.x;
    const int hi  = l >> 4;     // which half-wave
    const int amp = l & 15;     // basis-state index (4 qubits)
    const int gA  = hi;         // gate handled in register set A (0 or 1)
    const int gB  = 2 + hi;     // gate handled in register set B (2 or 3)

    // constant input x = qcnn_w + qcnn_b (features are all-ones)
    float x[HID];
#pragma unroll
    for (int h = 0; h < HID; ++h) x[h] = qcnn_w[h] + qcnn_b[h];

    // precompute: constant part of gate pre-activations, hx weights,
    // and cos/sin of trainable RX params (theta/2)
    float cpA[NQ], cpB[NQ];
    float lwA[HID][NQ], lwB[HID][NQ];
    float pcA[NQ], psA[NQ], pcB[NQ], psB[NQ];
#pragma unroll
    for (int w = 0; w < NQ; ++w) {
        float sa = lin_b[gA * NQ + w];
        float sb = lin_b[gB * NQ + w];
#pragma unroll
        for (int j = 0; j < HID; ++j) {
            sa += x[j] * lin_w[gA * 2 * HID * NQ + j * NQ + w];
            sb += x[j] * lin_w[gB * 2 * HID * NQ + j * NQ + w];
            lwA[j][w] = lin_w[gA * 2 * HID * NQ + (HID + j) * NQ + w];
            lwB[j][w] = lin_w[gB * 2 * HID * NQ + (HID + j) * NQ + w];
        }
        cpA[w] = sa; cpB[w] = sb;
        float ta = 0.5f * rx_params[gA * NQ + w];
        float tb = 0.5f * rx_params[gB * NQ + w];
        pcA[w] = __cosf(ta); psA[w] = __sinf(ta);
        pcB[w] = __cosf(tb); psB[w] = __sinf(tb);
    }

    float hx[NQ] = {0.f, 0.f, 0.f, 0.f};
    float cx[NQ] = {0.f, 0.f, 0.f, 0.f};

    for (int t = 0; t < T_SEQ; ++t) {
        // data-encoded RX angles (theta/2) for this step
        float thA[NQ], thB[NQ];
#pragma unroll
        for (int w = 0; w < NQ; ++w) {
            float sa = cpA[w], sb = cpB[w];
#pragma unroll
            for (int j = 0; j < HID; ++j) {
                sa = __fmaf_rn(hx[j], lwA[j][w], sa);
                sb = __fmaf_rn(hx[j], lwB[j][w], sb);
            }
            thA[w] = 0.5f * sa;
            thB[w] = 0.5f * sb;
        }

        // |0...0> initial state
        float rA = (amp == 0) ? 1.0f : 0.0f, iA = 0.0f;
        float rB = rA,                        iB = 0.0f;

        // data RX layer
#pragma unroll
        for (int w = 0; w < NQ; ++w) {
            rx_apply(rA, iA, __cosf(thA[w]), __sinf(thA[w]), 1 << w);
            rx_apply(rB, iB, __cosf(thB[w]), __sinf(thB[w]), 1 << w);
        }
        // trainable RX layer
#pragma unroll
        for (int w = 0; w < NQ; ++w) {
            rx_apply(rA, iA, pcA[w], psA[w], 1 << w);
            rx_apply(rB, iB, pcB[w], psB[w], 1 << w);
        }
        // CNOT ring 0->1->2->3->0
#pragma unroll
        for (int w = 0; w < NQ; ++w) {
            cnot_apply(rA, iA, amp, w, (w + 1) & 3);
            cnot_apply(rB, iB, amp, w, (w + 1) & 3);
        }

        // measure <Z_w> for each wire, then LSTM cell update
        float pA = rA * rA + iA * iA;
        float pB = rB * rB + iB * iB;
#pragma unroll
        for (int w = 0; w < NQ; ++w) {
            float sA = ((amp >> w) & 1) ? -pA : pA;
            float sB = ((amp >> w) & 1) ? -pB : pB;
            sA = half_reduce(sA);          // z of gate gA, uniform in half
            sB = half_reduce(sB);          // z of gate gB
            float oA = __shfl_xor(sA, 16, 32);
            float oB = __shfl_xor(sB, 16, 32);
            float z0 = hi ? oA : sA;       // gate 0: forget
            float z1 = hi ? sA : oA;       // gate 1: input
            float z2 = hi ? oB : sB;       // gate 2: update
            float z3 = hi ? sB : oB;       // gate 3: output
            float fg = fast_sigmoid(z0);
            float ig = fast_sigmoid(z1);
            float gg = fast_tanh(z2);
            float og = fast_sigmoid(z3);
            cx[w] = __fmaf_rn(fg, cx[w], ig * gg);
            hx[w] = og * fast_tanh(cx[w]);
        }

        if (l == 0) {
            float4 o = make_float4(hx[0], hx[1], hx[2], hx[3]);
            *reinterpret_cast<float4*>(hx_out + (size_t)t * HID) = o;
        }
    }
}

// ---------------------------------------------------------------------------
// Kernel 2: per-row softmax statistics.  One 256-thread block per row.
// logit[c] = tag_b[c] + hx . tag_w[:,c];  lse[t] = max + log(sum exp(.-max))
// tag_w (512KB) + tag_b are L2-resident, so the two passes are L2 traffic.
// ---------------------------------------------------------------------------
__global__ __launch_bounds__(256) void row_stats_kernel(
    const float* __restrict__ hx, const float* __restrict__ tag_w,
    const float* __restrict__ tag_b, float* __restrict__ lse)
{
    __shared__ float red[256];
    const int t   = blockIdx.x;
    const int tid = threadIdx.x;
    const float h0 = hx[t * 4 + 0], h1 = hx[t * 4 + 1];
    const float h2 = hx[t * 4 + 2], h3 = hx[t * 4 + 3];

    float m = -INFINITY;
    for (int c = tid; c < N_TAGS; c += 256) {
        float lg = tag_b[c];
        lg = __fmaf_rn(h0, tag_w[c],             lg);
        lg = __fmaf_rn(h1, tag_w[N_TAGS + c],    lg);
        lg = __fmaf_rn(h2, tag_w[2 * N_TAGS + c], lg);
        lg = __fmaf_rn(h3, tag_w[3 * N_TAGS + c], lg);
        m = fmaxf(m, lg);
    }
    red[tid] = m; __syncthreads();
    for (int s = 128; s > 0; s >>= 1) {
        if (tid < s) red[tid] = fmaxf(red[tid], red[tid + s]);
        __syncthreads();
    }
    m = red[0]; __syncthreads();

    float sum = 0.f;
    for (int c = tid; c < N_TAGS; c += 256) {
        float lg = tag_b[c];
        lg = __fmaf_rn(h0, tag_w[c],             lg);
        lg = __fmaf_rn(h1, tag_w[N_TAGS + c],    lg);
        lg = __fmaf_rn(h2, tag_w[2 * N_TAGS + c], lg);
        lg = __fmaf_rn(h3, tag_w[3 * N_TAGS + c], lg);
        sum += __expf(lg - m);
    }
    red[tid] = sum; __syncthreads();
    for (int s = 128; s > 0; s >>= 1) {
        if (tid < s) red[tid] += red[tid + s];
        __syncthreads();
    }
    if (tid == 0) lse[t] = m + __logf(red[0]);
}

// ---------------------------------------------------------------------------
// Kernel 3: output GEMM + log-softmax finalize via V_WMMA_F32_16X16X4_F32.
// One wave per 16x16 output tile; C is pre-loaded with (tag_b[col]-lse[row])
// so D = A*B + C is the final answer.  Store-bandwidth bound (524 MB).
// A layout (16x4 f32): lane m=l&15, VGPR0=K(2*hi), VGPR1=K(2*hi+1).
// B layout (4x16 f32): lane n=l&15, VGPR0=row K(2*hi), VGPR1=row K(2*hi+1).
// C/D layout: VGPR v -> row (v + 8*hi), col = l&15.
// ---------------------------------------------------------------------------
__global__ __launch_bounds__(256) void tagger_wmma_kernel(
    const float* __restrict__ hx, const float* __restrict__ tag_w,
    const float* __restrict__ tag_b, const float* __restrict__ lse,
    float* __restrict__ out)
{
    const int lane    = threadIdx.x & 31;
    const int wave    = threadIdx.x >> 5;
    const int colTile = blockIdx.x;                 // 0..1999
    const int rowTile = blockIdx.y * 8 + wave;      // 0..255
    const int hi  = lane >> 4;
    const int lo  = lane & 15;
    const int col = colTile * 16 + lo;
    const int k0  = 2 * hi;
    const int row0 = rowTile * 16;

    v2f a;
    a.x = hx[(row0 + lo) * 4 + k0];
    a.y = hx[(row0 + lo) * 4 + k0 + 1];

    v2f b;
    b.x = tag_w[(size_t)k0 * N_TAGS + col];
    b.y = tag_w[(size_t)(k0 + 1) * N_TAGS + col];

    const float tb = tag_b[col];
    v8f c;
#pragma unroll
    for (int v = 0; v < 8; ++v)
        c[v] = tb - lse[row0 + v + 8 * hi];

    // D = A(16x4) * B(4x16) + C ; emits v_wmma_f32_16x16x4_f32
    c = __builtin_amdgcn_wmma_f32_16x16x4_f32(
        /*neg_a=*/false, a, /*neg_b=*/false, b,
        /*c_mod=*/(short)0, c, /*reuse_a=*/false, /*reuse_b=*/false);

#pragma unroll
    for (int v = 0; v < 8; ++v)
        out[(size_t)(row0 + v + 8 * hi) * N_TAGS + col] = c[v];
}

// ---------------------------------------------------------------------------
// launch
// ---------------------------------------------------------------------------
extern "C" void kernel_launch(void* const* d_in, const int* in_sizes, int n_in,
                              void* d_out, int out_size, void* d_ws, size_t ws_size,
                              hipStream_t stream) {
    // setup_inputs order: sentence, embedding, qcnn_w, qcnn_b, lin_w, lin_b,
    //                     rx_params, tag_w, tag_b   (sentence/embedding unused)
    const float* qcnn_w    = (const float*)d_in[2];
    const float* qcnn_b    = (const float*)d_in[3];
    const float* lin_w     = (const float*)d_in[4];
    const float* lin_b     = (const float*)d_in[5];
    const float* rx_params = (const float*)d_in[6];
    const float* tag_w     = (const float*)d_in[7];
    const float* tag_b     = (const float*)d_in[8];
    float* out = (float*)d_out;

    float* hx_ws  = (float*)d_ws;                                        // [4096][4]
    float* lse_ws = (float*)((char*)d_ws + (size_t)T_SEQ * HID * sizeof(float)); // [4096]

    qlstm_recurrence_kernel<<<1, 32, 0, stream>>>(
        qcnn_w, qcnn_b, lin_w, lin_b, rx_params, hx_ws);

    row_stats_kernel<<<T_SEQ, 256, 0, stream>>>(hx_ws, tag_w, tag_b, lse_ws);

    dim3 grid3(N_TAGS / 16, T_SEQ / 16 / 8);   // (2000, 32), 8 waves/block
    tagger_wmma_kernel<<<grid3, 256, 0, stream>>>(hx_ws, tag_w, tag_b, lse_ws, out);
}